// CRF_63488206570022
// MI455X (gfx1250) — compile-verified
//
#include <hip/hip_runtime.h>

#define Bb 512
#define Ss 1024
#define Tt 64

typedef __attribute__((ext_vector_type(2))) float v2f;
typedef __attribute__((ext_vector_type(4))) float v4f;
typedef __attribute__((ext_vector_type(8))) float v8f;

// V-buffer row stride (floats): 68 keeps ds_load_b64 A-fragment fetches
// 8-byte aligned and bank-conflict-free on 64 banks.
#define VSTRIDE 68

// Raw-hardware transcendentals: inputs here never need the denormal fixup
// that __logf/__expf carry (log args >= ~0.9; exp args <= 0 underflow to 0).
static __device__ __forceinline__ float fast_exp(float x) {
  return __builtin_amdgcn_exp2f(x * 1.44269504088896340736f);  // v_exp_f32
}
static __device__ __forceinline__ float fast_log(float x) {
  return 0.69314718055994530942f * __builtin_amdgcn_logf(x);   // v_log_f32
}

// ---------------------------------------------------------------------------
// Forward (partition) kernel.
// Block = 128 threads = 4 wave32s; block owns 16 batch rows, wave n owns
// output columns [16n, 16n+16). dp lives in the WMMA f32 16x16 C/D layout:
//   element (VGPR v, lane L): row = v + 8*(L>=16), col = 16n + (L%16).
// Each step:  dp' = mask*( e_t + m + log( exp(dp-m) @ exp(trans) ) ) + (1-mask)*dp
// The 512x64 @ 64x64 f32 GEMM slice is 16 chained V_WMMA_F32_16X16X4_F32.
// ---------------------------------------------------------------------------
__global__ __launch_bounds__(128) void crf_forward_kernel(
    const float* __restrict__ em, const float* __restrict__ mask,
    const float* __restrict__ tr, float* __restrict__ z)
{
  __shared__ float smem[16 * VSTRIDE + 64 + 64];
  float* Vb   = smem;                 // 16 x 68 staging of V = exp(dp - m)
  float* maxb = smem + 16 * VSTRIDE;  // 16 rows x 4 wave-partials (row max)
  float* sumb = maxb + 64;            // 16 rows x 4 wave-partials (final sums)

  const int tid  = threadIdx.x;
  const int n    = tid >> 5;   // wave id -> N-tile
  const int lane = tid & 31;
  const int half = lane >> 4;
  const int l    = lane & 15;
  const int b0   = blockIdx.x * 16;
  const int col  = 16 * n + l;

  // B-fragments of W = exp(transitions): 4x16 f32 per k-step.
  // VGPR j holds K = 4k + j (lanes 0-15) / K = 4k + j + 2 (lanes 16-31).
  v2f w[16];
  #pragma unroll
  for (int k = 0; k < 16; ++k) {
    w[k].x = fast_exp(tr[(4 * k + 0 + 2 * half) * Tt + col]);
    w[k].y = fast_exp(tr[(4 * k + 1 + 2 * half) * Tt + col]);
  }

  // Per-(v) row base pointers for emissions / mask in the D layout.
  const float* eptr[8];
  const float* mptr[8];
  #pragma unroll
  for (int v = 0; v < 8; ++v) {
    const int gb = b0 + v + 8 * half;
    eptr[v] = em + (size_t)gb * Ss * Tt + col;
    mptr[v] = mask + (size_t)gb * Ss;
  }

  v8f dp = {0.f, 0.f, 0.f, 0.f, 0.f, 0.f, 0.f, 0.f};

  for (int s = 0; s < Ss; ++s) {
    // ---- per-wave partial row max (16-lane xor-tree within each half) ----
    float pm[8];
    #pragma unroll
    for (int v = 0; v < 8; ++v) {
      float x = dp[v];
      x = fmaxf(x, __shfl_xor(x, 1, 32));
      x = fmaxf(x, __shfl_xor(x, 2, 32));
      x = fmaxf(x, __shfl_xor(x, 4, 32));
      x = fmaxf(x, __shfl_xor(x, 8, 32));
      pm[v] = x;
    }
    if (l == 0) {  // lanes 0 and 16 publish rows 0-7 / 8-15
      #pragma unroll
      for (int v = 0; v < 8; ++v) maxb[(v + 8 * half) * 4 + n] = pm[v];
    }
    __syncthreads();

    // ---- combined row max; stage V = exp(dp - m) into LDS ----
    float m[8];
    #pragma unroll
    for (int v = 0; v < 8; ++v) {
      v4f q = *(const v4f*)(maxb + (v + 8 * half) * 4);
      float mm = fmaxf(fmaxf(q.x, q.y), fmaxf(q.z, q.w));
      m[v] = mm;
      Vb[(v + 8 * half) * VSTRIDE + col] = fast_exp(dp[v] - mm);
    }
    __syncthreads();

    // ---- acc(16x16) = V(16x64) @ W(64x16): 16 chained f32 WMMAs ----
    v8f acc = {0.f, 0.f, 0.f, 0.f, 0.f, 0.f, 0.f, 0.f};
    #pragma unroll
    for (int k = 0; k < 16; ++k) {
      // A frag: lanes 0-15 -> V[l][4k..4k+1], lanes 16-31 -> V[l][4k+2..4k+3]
      v2f a = *(const v2f*)(Vb + l * VSTRIDE + 4 * k + 2 * half);
      acc = __builtin_amdgcn_wmma_f32_16x16x4_f32(
          /*neg_a=*/false, a, /*neg_b=*/false, w[k],
          /*c_mod=*/(short)0, acc, /*reuse_a=*/false, /*reuse_b=*/false);
    }

    // ---- dp update in D layout ----
    #pragma unroll
    for (int v = 0; v < 8; ++v) {
      const float e  = eptr[v][(size_t)s * Tt];
      const float mt = mptr[v][s];
      const float nd = e + m[v] + fast_log(acc[v]);
      dp[v] = mt * nd + (1.f - mt) * dp[v];
    }
  }

  // ---- z[b] = logsumexp over the 64 columns of dp ----
  float pm[8];
  #pragma unroll
  for (int v = 0; v < 8; ++v) {
    float x = dp[v];
    x = fmaxf(x, __shfl_xor(x, 1, 32));
    x = fmaxf(x, __shfl_xor(x, 2, 32));
    x = fmaxf(x, __shfl_xor(x, 4, 32));
    x = fmaxf(x, __shfl_xor(x, 8, 32));
    pm[v] = x;
  }
  if (l == 0) {
    #pragma unroll
    for (int v = 0; v < 8; ++v) maxb[(v + 8 * half) * 4 + n] = pm[v];
  }
  __syncthreads();
  #pragma unroll
  for (int v = 0; v < 8; ++v) {
    v4f q = *(const v4f*)(maxb + (v + 8 * half) * 4);
    float mm = fmaxf(fmaxf(q.x, q.y), fmaxf(q.z, q.w));
    float ve = fast_exp(dp[v] - mm);
    ve += __shfl_xor(ve, 1, 32);
    ve += __shfl_xor(ve, 2, 32);
    ve += __shfl_xor(ve, 4, 32);
    ve += __shfl_xor(ve, 8, 32);
    pm[v] = ve;  // reuse as partial sums
  }
  if (l == 0) {
    #pragma unroll
    for (int v = 0; v < 8; ++v) sumb[(v + 8 * half) * 4 + n] = pm[v];
  }
  __syncthreads();
  if (n == 0 && lane < 16) {
    v4f qm = *(const v4f*)(maxb + lane * 4);
    v4f qs = *(const v4f*)(sumb + lane * 4);
    float mm = fmaxf(fmaxf(qm.x, qm.y), fmaxf(qm.z, qm.w));
    float ss = qs.x + qs.y + qs.z + qs.w;
    z[b0 + lane] = mm + fast_log(ss);
  }
}

// ---------------------------------------------------------------------------
// Gold path scores: scores[b] = sum_{s=1..S-1} (em[b,s,tag]+trans[tp,tc])*mask
// ---------------------------------------------------------------------------
__global__ __launch_bounds__(128) void crf_scores_kernel(
    const float* __restrict__ em, const int* __restrict__ tags,
    const float* __restrict__ mask, const float* __restrict__ tr,
    float* __restrict__ scores)
{
  const int b = blockIdx.x;
  const int tid = threadIdx.x;
  float acc = 0.f;
  for (int s = 1 + tid; s < Ss; s += 128) {
    const int tc = tags[b * Ss + s];
    const int tp = tags[b * Ss + s - 1];
    const float e = em[((size_t)b * Ss + s) * Tt + tc];
    acc += (e + tr[tp * Tt + tc]) * mask[b * Ss + s];
  }
  acc += __shfl_xor(acc, 1, 32);
  acc += __shfl_xor(acc, 2, 32);
  acc += __shfl_xor(acc, 4, 32);
  acc += __shfl_xor(acc, 8, 32);
  acc += __shfl_xor(acc, 16, 32);
  __shared__ float sb[4];
  if ((tid & 31) == 0) sb[tid >> 5] = acc;
  __syncthreads();
  if (tid == 0) scores[b] = sb[0] + sb[1] + sb[2] + sb[3];
}

// ---------------------------------------------------------------------------
// loss = -(1/B) * sum_b (z[b] - scores[b])
// ---------------------------------------------------------------------------
__global__ __launch_bounds__(512) void crf_finalize_kernel(
    const float* __restrict__ z, const float* __restrict__ scores,
    float* __restrict__ out)
{
  const int tid = threadIdx.x;  // 512 threads, one per batch row
  float v = z[tid] - scores[tid];
  v += __shfl_xor(v, 1, 32);
  v += __shfl_xor(v, 2, 32);
  v += __shfl_xor(v, 4, 32);
  v += __shfl_xor(v, 8, 32);
  v += __shfl_xor(v, 16, 32);
  __shared__ float sb[16];
  if ((tid & 31) == 0) sb[tid >> 5] = v;
  __syncthreads();
  if (tid < 16) {
    float x = sb[tid];
    x += __shfl_xor(x, 1, 32);
    x += __shfl_xor(x, 2, 32);
    x += __shfl_xor(x, 4, 32);
    x += __shfl_xor(x, 8, 32);
    if (tid == 0) out[0] = -x * (1.f / (float)Bb);
  }
}

extern "C" void kernel_launch(void* const* d_in, const int* in_sizes, int n_in,
                              void* d_out, int out_size, void* d_ws, size_t ws_size,
                              hipStream_t stream) {
  (void)in_sizes; (void)n_in; (void)out_size; (void)ws_size;
  const float* em   = (const float*)d_in[0];
  const int*   tags = (const int*)d_in[1];
  const float* mask = (const float*)d_in[2];
  const float* tr   = (const float*)d_in[3];
  float* out = (float*)d_out;

  float* z      = (float*)d_ws;   // B floats
  float* scores = z + Bb;         // B floats

  crf_forward_kernel<<<Bb / 16, 128, 0, stream>>>(em, mask, tr, z);
  crf_scores_kernel<<<Bb, 128, 0, stream>>>(em, tags, mask, tr, scores);
  crf_finalize_kernel<<<1, 512, 0, stream>>>(z, scores, out);
}